// SplineGCN_60155311948259
// MI455X (gfx1250) — compile-verified
//
#include <hip/hip_runtime.h>
#include <hip/hip_bf16.h>

#define NFEAT 128
#define HID   16

typedef float v2f __attribute__((ext_vector_type(2)));
typedef float v8f __attribute__((ext_vector_type(8)));

__device__ __forceinline__ float eluf(float v) {
    return v > 0.0f ? v : (expf(v) - 1.0f);
}

// ---------------------------------------------------------------------------
// Layer-1 node transform: for each 16-node tile compute
//   h0 = x @ w1[0], h1 = x @ w1[1], rp = x @ root1 + b1        (all [16,16])
// One wave per tile, three f32 WMMA accumulators, K=128 swept in steps of 4
// with V_WMMA_F32_16X16X4_F32. B fragments (shared by all waves/tiles) are
// pre-swizzled into LDS once per block so the hot loop is
//   global_load_b64 (A) + 3x ds_load_b64 (B) + 3x v_wmma.
// ---------------------------------------------------------------------------
__global__ void __launch_bounds__(256)
spline_l1_node(const float* __restrict__ x,
               const float* __restrict__ w0, const float* __restrict__ w1b,
               const float* __restrict__ root, const float* __restrict__ bias,
               float* __restrict__ h0, float* __restrict__ h1,
               float* __restrict__ rp, int n_tiles)
{
    __shared__ v2f bf[3][32][32];               // [matrix][kstep][lane], 24 KB

    // --- stage pre-swizzled B fragments (all 256 threads) ---
    for (int i = threadIdx.x; i < 3 * 32 * 32; i += 256) {
        const int mat   = i >> 10;              // /1024
        const int kstep = (i >> 5) & 31;
        const int ln    = i & 31;
        const int col   = ln & 15;
        const int kb    = kstep * 4 + ((ln >> 4) << 1);
        const float* w  = (mat == 0) ? w0 : (mat == 1) ? w1b : root;
        v2f b;
        b.x = w[(kb    ) * HID + col];
        b.y = w[(kb + 1) * HID + col];
        bf[mat][kstep][ln] = b;
    }
    __syncthreads();

    const int wave = threadIdx.x >> 5;
    const int lane = threadIdx.x & 31;
    const int tile = blockIdx.x * 8 + wave;
    if (tile >= n_tiles) return;                // wave-uniform; barrier already passed

    const int mrow  = lane & 15;
    const int khalf = (lane >> 4) << 1;         // lanes 16..31 hold K+2,K+3
    const int row0  = tile << 4;
    const float* xr = x + (size_t)(row0 + mrow) * NFEAT + khalf;

    v8f c0 = {0,0,0,0,0,0,0,0};
    v8f c1 = {0,0,0,0,0,0,0,0};
    v8f cr = {0,0,0,0,0,0,0,0};

#pragma unroll 8
    for (int s = 0; s < 32; ++s) {
        v2f a  = *(const v2f*)(xr + s * 4);     // A: x[row, kb], x[row, kb+1]
        v2f b0 = bf[0][s][lane];
        v2f b1 = bf[1][s][lane];
        v2f br = bf[2][s][lane];
        c0 = __builtin_amdgcn_wmma_f32_16x16x4_f32(false, a, false, b0, (short)0, c0, false, false);
        c1 = __builtin_amdgcn_wmma_f32_16x16x4_f32(false, a, false, b1, (short)0, c1, false, false);
        cr = __builtin_amdgcn_wmma_f32_16x16x4_f32(false, a, false, br, (short)0, cr, false, false);
    }

    const float bb    = bias[mrow];
    const int   mbase = (lane >> 4) << 3;       // C/D: vgpr r -> M = r + 8*(lane>=16)
#pragma unroll
    for (int r = 0; r < 8; ++r) {
        const size_t idx = (size_t)(row0 + mbase + r) * HID + mrow;
        h0[idx] = c0[r];
        h1[idx] = c1[r];
        rp[idx] = cr[r] + bb;
    }
}

// ---------------------------------------------------------------------------
// Layer-2 node transform, fused with layer-1 finalize:
//   h = elu(agg/max(deg,1) + rp1)   built on the fly into WMMA A-fragments
//   g0 = h @ w2[0], g1 = h @ w2[1], rp2 = h @ root2 + b2
// ---------------------------------------------------------------------------
__global__ void __launch_bounds__(256)
spline_l2_node(const float* __restrict__ agg, const float* __restrict__ deg,
               const float* __restrict__ rp1,
               const float* __restrict__ w0, const float* __restrict__ w1b,
               const float* __restrict__ root, const float* __restrict__ bias,
               float* __restrict__ g0, float* __restrict__ g1,
               float* __restrict__ rp2, int n_tiles)
{
    __shared__ v2f bf[3][4][32];                // 3 KB

    for (int i = threadIdx.x; i < 3 * 4 * 32; i += 256) {
        const int mat   = i >> 7;               // /128
        const int kstep = (i >> 5) & 3;
        const int ln    = i & 31;
        const int col   = ln & 15;
        const int kb    = kstep * 4 + ((ln >> 4) << 1);
        const float* w  = (mat == 0) ? w0 : (mat == 1) ? w1b : root;
        v2f b;
        b.x = w[(kb    ) * HID + col];
        b.y = w[(kb + 1) * HID + col];
        bf[mat][kstep][ln] = b;
    }
    __syncthreads();

    const int wave = threadIdx.x >> 5;
    const int lane = threadIdx.x & 31;
    const int tile = blockIdx.x * 8 + wave;
    if (tile >= n_tiles) return;

    const int mrow  = lane & 15;
    const int khalf = (lane >> 4) << 1;
    const int row0  = tile << 4;
    const int row   = row0 + mrow;

    const float inv = 1.0f / fmaxf(deg[row], 1.0f);
    const size_t rbase = (size_t)row * HID;

    v8f c0 = {0,0,0,0,0,0,0,0};
    v8f c1 = {0,0,0,0,0,0,0,0};
    v8f cr = {0,0,0,0,0,0,0,0};

#pragma unroll
    for (int s = 0; s < 4; ++s) {
        const int kb = s * 4 + khalf;
        v2f a;
        a.x = eluf(agg[rbase + kb    ] * inv + rp1[rbase + kb    ]);
        a.y = eluf(agg[rbase + kb + 1] * inv + rp1[rbase + kb + 1]);
        v2f b0 = bf[0][s][lane];
        v2f b1 = bf[1][s][lane];
        v2f br = bf[2][s][lane];
        c0 = __builtin_amdgcn_wmma_f32_16x16x4_f32(false, a, false, b0, (short)0, c0, false, false);
        c1 = __builtin_amdgcn_wmma_f32_16x16x4_f32(false, a, false, b1, (short)0, c1, false, false);
        cr = __builtin_amdgcn_wmma_f32_16x16x4_f32(false, a, false, br, (short)0, cr, false, false);
    }

    const float bb    = bias[mrow];
    const int   mbase = (lane >> 4) << 3;
#pragma unroll
    for (int r = 0; r < 8; ++r) {
        const size_t idx = (size_t)(row0 + mbase + r) * HID + mrow;
        g0[idx]  = c0[r];
        g1[idx]  = c1[r];
        rp2[idx] = cr[r] + bb;
    }
}

// ---------------------------------------------------------------------------
// Edge scatter: 16 threads per edge (one per hidden feature) so the 64B rows
// of h0[src]/h1[src] are coalesced; f32 atomic adds into agg[dst], lane 0
// counts degree on the first pass. h0/h1 (3.2MB each) stay L2-resident.
// ---------------------------------------------------------------------------
__global__ void __launch_bounds__(256)
edge_scatter(const int* __restrict__ src, const int* __restrict__ dst,
             const float* __restrict__ u,
             const float* __restrict__ h0, const float* __restrict__ h1,
             float* __restrict__ agg, float* __restrict__ deg,
             int n_edges, int do_deg)
{
    const long long t = (long long)blockIdx.x * blockDim.x + threadIdx.x;
    const long long e = t >> 4;
    const int       o = (int)(t & 15);
    if (e >= n_edges) return;
    const int   s  = src[e];
    const int   d  = dst[e];
    const float uu = u[e];
    const size_t sb = (size_t)s * HID + o;
    const float m = (1.0f - uu) * h0[sb] + uu * h1[sb];
    atomicAdd(&agg[(size_t)d * HID + o], m);
    if (do_deg && o == 0) atomicAdd(&deg[d], 1.0f);
}

// ---------------------------------------------------------------------------
// Final: out = log_softmax(agg/max(deg,1) + rp2), one thread per node.
// ---------------------------------------------------------------------------
__global__ void __launch_bounds__(256)
finalize_logsoftmax(const float* __restrict__ agg, const float* __restrict__ deg,
                    const float* __restrict__ rp,
                    float* __restrict__ out, int n_nodes)
{
    const int nid = blockIdx.x * blockDim.x + threadIdx.x;
    if (nid >= n_nodes) return;
    const float inv = 1.0f / fmaxf(deg[nid], 1.0f);
    const size_t base = (size_t)nid * HID;
    float v[HID];
    float mx = -__builtin_inff();
#pragma unroll
    for (int o = 0; o < HID; ++o) {
        v[o] = agg[base + o] * inv + rp[base + o];
        mx = fmaxf(mx, v[o]);
    }
    float s = 0.0f;
#pragma unroll
    for (int o = 0; o < HID; ++o) s += expf(v[o] - mx);
    const float lse = logf(s);
#pragma unroll
    for (int o = 0; o < HID; ++o) out[base + o] = v[o] - mx - lse;
}

__global__ void __launch_bounds__(256)
zero_f32(float* __restrict__ p, long long n)
{
    const long long i = (long long)blockIdx.x * blockDim.x + threadIdx.x;
    if (i < n) p[i] = 0.0f;
}

extern "C" void kernel_launch(void* const* d_in, const int* in_sizes, int n_in,
                              void* d_out, int out_size, void* d_ws, size_t ws_size,
                              hipStream_t stream)
{
    const float* x     = (const float*)d_in[0];
    const int*   ei    = (const int*)  d_in[1];   // [2, E]
    const float* u     = (const float*)d_in[2];   // [E, 1]
    const float* w1    = (const float*)d_in[3];   // [2,128,16]
    const float* root1 = (const float*)d_in[4];   // [128,16]
    const float* b1    = (const float*)d_in[5];   // [16]
    const float* w2    = (const float*)d_in[6];   // [2,16,16]
    const float* root2 = (const float*)d_in[7];   // [16,16]
    const float* b2    = (const float*)d_in[8];   // [16]
    float*       out   = (float*)d_out;

    const int n_nodes = in_sizes[0] / NFEAT;      // 50000 (multiple of 16)
    const int n_edges = in_sizes[1] / 2;          // 800000
    const int n_tiles = n_nodes / 16;

    const int* src = ei;
    const int* dst = ei + n_edges;

    // workspace layout (floats): 5*N*16 + N  ~= 16.2 MB
    float* ws  = (float*)d_ws;
    const size_t nh = (size_t)n_nodes * HID;
    float* h0  = ws;            // layer1 h0  / layer2 g0
    float* h1  = h0  + nh;      // layer1 h1  / layer2 g1
    float* rp1 = h1  + nh;      // root-part layer 1 (incl bias)
    float* rp2 = rp1 + nh;      // root-part layer 2 (incl bias)
    float* agg = rp2 + nh;      // scatter accumulator (reused both layers)
    float* deg = agg + nh;      // degree (computed once, reused)

    const dim3 blk(256);
    const dim3 grid_node((n_tiles + 7) / 8);
    const dim3 grid_edge((unsigned)(((long long)n_edges * 16 + 255) / 256));
    const dim3 grid_nelem((unsigned)((nh + 255) / 256));
    const dim3 grid_zero1((unsigned)((nh + (size_t)n_nodes + 255) / 256));
    const dim3 grid_fin((n_nodes + 255) / 256);

    // Layer 1
    spline_l1_node<<<grid_node, blk, 0, stream>>>(x, w1, w1 + NFEAT * HID, root1, b1,
                                                  h0, h1, rp1, n_tiles);
    zero_f32<<<grid_zero1, blk, 0, stream>>>(agg, (long long)nh + n_nodes); // agg + deg
    edge_scatter<<<grid_edge, blk, 0, stream>>>(src, dst, u, h0, h1, agg, deg, n_edges, 1);

    // Layer 2 (finalize-1 + ELU fused into A-fragment construction)
    spline_l2_node<<<grid_node, blk, 0, stream>>>(agg, deg, rp1, w2, w2 + HID * HID,
                                                  root2, b2, h0, h1, rp2, n_tiles);
    zero_f32<<<grid_nelem, blk, 0, stream>>>(agg, (long long)nh);
    edge_scatter<<<grid_edge, blk, 0, stream>>>(src, dst, u, h0, h1, agg, deg, n_edges, 0);

    // Finalize + log_softmax
    finalize_logsoftmax<<<grid_fin, blk, 0, stream>>>(agg, deg, rp2, out, n_nodes);
}